// Set2SetLoss_20160576487954
// MI455X (gfx1250) — compile-verified
//
#include <hip/hip_runtime.h>
#include <math.h>

// Set2Set margin loss, K=128 groups x N=15 embeddings x DIM=512, fp32.
// Strategy: native fp32 WMMA (v_wmma_f32_16x16x4_f32) per 16x16 tile
// (15x15 block padded via clamped row index, masked in epilogue).

typedef float v2f __attribute__((ext_vector_type(2)));
typedef float v8f __attribute__((ext_vector_type(8)));

#define KBLK   128
#define NPER   15
#define DIM    512
#define NPAIR  8128   // 128*127/2

// workspace layout (float offsets)
#define WS_SQN   0                       // 1920 row squared norms
#define WS_DAA   2048                    // 128 * 256 within-block dist tiles (16x16 padded)
#define WS_PAIR  (WS_DAA + KBLK * 256)   // 8128 per-pair margin partial sums
#define WS_MDIST (WS_PAIR + NPAIR)       // 128 per-group mean centroid distances

// ---------------------------------------------------------------------------
// Row squared norms: one wave per row, 8 rows per 256-thread block.
__global__ void __launch_bounds__(256) sqn_kernel(const float* __restrict__ E,
                                                  float* __restrict__ ws) {
    const int wid  = threadIdx.x >> 5;
    const int lane = threadIdx.x & 31;
    const int row  = blockIdx.x * 8 + wid;
    const float* p = E + (long)row * DIM;
    float s = 0.f;
#pragma unroll
    for (int t = 0; t < 4; ++t) {
        const float4 v = *(const float4*)(p + 4 * (lane + 32 * t));
        s += v.x * v.x + v.y * v.y + v.z * v.z + v.w * v.w;
    }
#pragma unroll
    for (int off = 16; off > 0; off >>= 1) s += __shfl_xor(s, off, 32);
    if (lane == 0) ws[WS_SQN + row] = s;
}

// ---------------------------------------------------------------------------
// 16x16 Gram tile  G = Ea(16x512) @ Eb(16x512)^T  via 128 fp32 WMMA steps.
// A-frag & B-frag share the same per-lane addressing: row = lane&15 (clamped
// to 14 for the padding row), cols k0 + 2*(lane>>4) + {0,1}.
__device__ __forceinline__ v8f gram_tile(const float* __restrict__ Ea,
                                         const float* __restrict__ Eb,
                                         int lane) {
    const int r  = lane & 15;
    const int h  = lane >> 4;
    const int rc = (r < NPER) ? r : (NPER - 1);   // clamp padded row (branchless, keeps EXEC full)
    const float* pa = Ea + rc * DIM + 2 * h;
    const float* pb = Eb + rc * DIM + 2 * h;
    v8f acc = {0.f, 0.f, 0.f, 0.f, 0.f, 0.f, 0.f, 0.f};
#pragma unroll 8
    for (int k0 = 0; k0 < DIM; k0 += 4) {
        v2f a = *(const v2f*)(pa + k0);
        v2f b = *(const v2f*)(pb + k0);
        acc = __builtin_amdgcn_wmma_f32_16x16x4_f32(
            /*neg_a=*/false, a, /*neg_b=*/false, b,
            /*c_mod=*/(short)0, acc, /*reuse_a=*/false, /*reuse_b=*/false);
    }
    return acc;
}

// ---------------------------------------------------------------------------
// Within-block distance tiles dAA(a): one wave per block, store 16x16 tile.
__global__ void __launch_bounds__(32) daa_kernel(const float* __restrict__ E,
                                                 float* __restrict__ ws) {
    const int a    = blockIdx.x;
    const int lane = threadIdx.x;
    const float* Ea = E + (long)a * NPER * DIM;
    v8f g = gram_tile(Ea, Ea, lane);

    const int n = lane & 15;
    const int h = lane >> 4;
    const float* sqn = ws + WS_SQN;
    const float nb = sqn[a * NPER + ((n < NPER) ? n : NPER - 1)];
    float* daa = ws + WS_DAA + a * 256;
#pragma unroll
    for (int v = 0; v < 8; ++v) {
        const int m = v + 8 * h;
        const float na = sqn[a * NPER + ((m < NPER) ? m : NPER - 1)];
        const float sq = na + nb - 2.f * g[v];
        daa[m * 16 + n] = (sq > 0.f) ? sqrtf(sq) : 0.f;   // _safe_dist
    }
}

// ---------------------------------------------------------------------------
// Cross-block pair (a<b): WMMA dAB tile + margin accumulation vs LDS-staged dAA(a).
__global__ void __launch_bounds__(32) pair_kernel(const float* __restrict__ E,
                                                  float* __restrict__ ws) {
    __shared__ float sdaa[256];
    const int lane = threadIdx.x;

    // p -> (a,b), a<b  (uniform scalar loop, <=127 iterations)
    int rem = blockIdx.x, a = 0;
    while (rem >= (KBLK - 1 - a)) { rem -= (KBLK - 1 - a); ++a; }
    const int b = a + 1 + rem;

    // stage dAA(a) tile into LDS (256 floats, 8 per lane)
    const float* daa = ws + WS_DAA + a * 256;
#pragma unroll
    for (int t = 0; t < 8; ++t) sdaa[lane * 8 + t] = daa[lane * 8 + t];

    const float* Ea = E + (long)a * NPER * DIM;
    const float* Eb = E + (long)b * NPER * DIM;
    v8f g = gram_tile(Ea, Eb, lane);
    __syncthreads();

    const int n = lane & 15;        // l column of dAB owned by this lane
    const int h = lane >> 4;
    const float* sqn = ws + WS_SQN;
    const float nb = sqn[b * NPER + ((n < NPER) ? n : NPER - 1)];

    float s = 0.f;
    if (n < NPER) {
#pragma unroll
        for (int v = 0; v < 8; ++v) {
            const int i = v + 8 * h;
            if (i < NPER) {
                const float na = sqn[a * NPER + i];
                const float sq = na + nb - 2.f * g[v];
                const float dab = (sq > 0.f) ? sqrtf(sq) : 0.f;
                for (int j = i + 1; j < NPER; ++j)
                    s += fmaxf(sdaa[i * 16 + j] - dab + 1.5f, 0.f);
            }
        }
    }
#pragma unroll
    for (int off = 16; off > 0; off >>= 1) s += __shfl_xor(s, off, 32);
    if (lane == 0) ws[WS_PAIR + blockIdx.x] = s;   // deterministic: no atomics
}

// ---------------------------------------------------------------------------
// Per-group mean distance to centroid: one wave per group, 16 dims per lane.
__global__ void __launch_bounds__(32) radius_kernel(const float* __restrict__ E,
                                                    float* __restrict__ ws) {
    const int k    = blockIdx.x;
    const int lane = threadIdx.x;
    const float* G = E + (long)k * NPER * DIM;

    float c[16];
#pragma unroll
    for (int t = 0; t < 16; ++t) c[t] = 0.f;
    for (int r = 0; r < NPER; ++r) {
#pragma unroll
        for (int t = 0; t < 16; ++t) c[t] += G[r * DIM + lane + 32 * t];
    }
#pragma unroll
    for (int t = 0; t < 16; ++t) c[t] *= (1.0f / NPER);

    float acc = 0.f;
    for (int r = 0; r < NPER; ++r) {
        float s = 0.f;
#pragma unroll
        for (int t = 0; t < 16; ++t) {
            const float d = G[r * DIM + lane + 32 * t] - c[t];
            s += d * d;
        }
#pragma unroll
        for (int off = 16; off > 0; off >>= 1) s += __shfl_xor(s, off, 32);
        acc += (s > 0.f) ? sqrtf(s) : 0.f;        // _safe_dist
    }
    if (lane == 0) ws[WS_MDIST + k] = acc * (1.0f / NPER);
}

// ---------------------------------------------------------------------------
// Final deterministic reductions -> scalar loss.
__global__ void __launch_bounds__(256) final_kernel(const float* __restrict__ ws,
                                                    float* __restrict__ out) {
    __shared__ float red[256];
    __shared__ float sh_margin, sh_total;
    const int tid = threadIdx.x;

    // margin partial sum over 8128 pairs
    float s = 0.f;
    for (int i = tid; i < NPAIR; i += 256) s += ws[WS_PAIR + i];
    red[tid] = s; __syncthreads();
    for (int st = 128; st > 0; st >>= 1) { if (tid < st) red[tid] += red[tid + st]; __syncthreads(); }
    if (tid == 0) sh_margin = red[0];
    __syncthreads();

    // total radius = mean(mean_dist)
    const float m = (tid < KBLK) ? ws[WS_MDIST + tid] : 0.f;
    red[tid] = m; __syncthreads();
    for (int st = 128; st > 0; st >>= 1) { if (tid < st) red[tid] += red[tid + st]; __syncthreads(); }
    if (tid == 0) sh_total = red[0] * (1.0f / KBLK);
    __syncthreads();

    // radius penalty = BETA * mean(|mean_dist/total - 1|)
    const float av = (tid < KBLK) ? fabsf(m / sh_total - 1.0f) : 0.f;
    red[tid] = av; __syncthreads();
    for (int st = 128; st > 0; st >>= 1) { if (tid < st) red[tid] += red[tid + st]; __syncthreads(); }
    if (tid == 0) {
        const float margin  = sh_margin / (1575.0f /* N*N*(N-1)/2 */) / (float)NPAIR;
        const float penalty = 0.05f * (red[0] * (1.0f / KBLK));
        out[0] = margin + penalty;
    }
}

// ---------------------------------------------------------------------------
extern "C" void kernel_launch(void* const* d_in, const int* in_sizes, int n_in,
                              void* d_out, int out_size, void* d_ws, size_t ws_size,
                              hipStream_t stream) {
    const float* E = (const float*)d_in[0];
    float* ws  = (float*)d_ws;
    float* out = (float*)d_out;

    sqn_kernel   <<<(KBLK * NPER) / 8, 256, 0, stream>>>(E, ws);
    daa_kernel   <<<KBLK,              32, 0, stream>>>(E, ws);
    pair_kernel  <<<NPAIR,             32, 0, stream>>>(E, ws);
    radius_kernel<<<KBLK,              32, 0, stream>>>(E, ws);
    final_kernel <<<1,                256, 0, stream>>>(ws, out);
}